// DilateAttentionToken_70282844831840
// MI455X (gfx1250) — compile-verified
//
#include <hip/hip_runtime.h>
#include <stdint.h>

// Problem constants (from reference).
#define N_TOK   4096
#define NCH     512
#define HD      32
#define NHEAD   16
#define KW      9
#define DIL     3
#define PAD     12
#define TILE    128          // tokens per block
#define KPITCH  160          // k/v tile row pitch in floats (40 x float4, covers 128+24 halo)
#define OPITCH  33           // out staging pitch (bank-conflict-free)
#define THREADS 128

__global__ __launch_bounds__(THREADS)
void dilated_attn_kernel(const float* __restrict__ q,
                         const float* __restrict__ k,
                         const float* __restrict__ v,
                         float* __restrict__ out)
{
    __shared__ float kT[HD * KPITCH];     // 20 KB
    __shared__ float vT[HD * KPITCH];     // 20 KB
    __shared__ float oT[TILE * OPITCH];   // 16.5 KB

    const int tid  = threadIdx.x;
    const int tilesPerSeq = N_TOK / TILE;            // 32
    const int bid  = blockIdx.x;
    const int tile = bid % tilesPerSeq;
    const int bh   = bid / tilesPerSeq;
    const int head = bh % NHEAD;
    const int b    = bh / NHEAD;
    const int n0   = tile * TILE;

    // Uniform (scalar) global base pointers for this (b, head).
    const size_t chanBase = (size_t)(b * NCH + head * HD) * N_TOK;
    const float* qbase = q + chanBase;
    const float* kbase = k + chanBase;
    const float* vbase = v + chanBase;
    float*       obase = out + (size_t)b * N_TOK * NCH + head * HD;

    // LDS byte addresses: low 32 bits of the flat shared pointer.
    unsigned ldsK = (unsigned)(uintptr_t)(void*)kT;
    unsigned ldsV = (unsigned)(uintptr_t)(void*)vT;
    unsigned ldsO = (unsigned)(uintptr_t)(void*)oT;

    // ---- Async DMA of k/v tiles into LDS (CDNA5 async-tensor path) ----
    // 32 rows x 40 float4 chunks per tile = 1280 chunk-copies, 10 per thread.
    // Tile columns cover global [n0-12, n0+148); edges clamped (masked later).
    #pragma unroll
    for (int it = 0; it < 10; ++it) {
        int t    = tid + it * THREADS;
        int row  = t / 40;
        int ch   = t - row * 40;
        int gcol = n0 - PAD + ch * 4;
        gcol = gcol < 0 ? 0 : (gcol > N_TOK - 4 ? N_TOK - 4 : gcol);
        unsigned voff = (unsigned)((row * N_TOK + gcol) * 4);
        unsigned ldk  = ldsK + (unsigned)((row * KPITCH + ch * 4) * 4);
        unsigned ldv  = ldsV + (unsigned)((row * KPITCH + ch * 4) * 4);
        asm volatile("global_load_async_to_lds_b128 %0, %1, %2"
                     :: "v"(ldk), "v"(voff), "s"(kbase) : "memory");
        asm volatile("global_load_async_to_lds_b128 %0, %1, %2"
                     :: "v"(ldv), "v"(voff), "s"(vbase) : "memory");
    }

    // ---- q straight to registers while the DMA is in flight ----
    // lane = token => consecutive addresses per dd row: fully coalesced.
    float qv[HD];
    const float* qcol = qbase + n0 + tid;
    #pragma unroll
    for (int dd = 0; dd < HD; ++dd) qv[dd] = qcol[dd * N_TOK];

    asm volatile("s_wait_asynccnt 0x0" ::: "memory");
    __syncthreads();

    // ---- scores: s[j] = sum_dd q[dd] * k[dd][tid + 3j] (tile-local) ----
    float s[KW];
    #pragma unroll
    for (int j = 0; j < KW; ++j) s[j] = 0.0f;

    #pragma unroll 8
    for (int dd = 0; dd < HD; ++dd) {
        const float qd = qv[dd];
        const float* krow = &kT[dd * KPITCH + tid];
        #pragma unroll
        for (int j = 0; j < KW; ++j)
            s[j] = fmaf(qd, krow[j * DIL], s[j]);
    }

    // Mask exactly like the reference: OOB positions get score == 0 (pre-softmax).
    const float scale = 0.17677669529663687f;   // 32^-0.5
    const int gb = n0 + tid - PAD;
    #pragma unroll
    for (int j = 0; j < KW; ++j) {
        bool ok = (unsigned)(gb + j * DIL) < (unsigned)N_TOK;
        s[j] = ok ? s[j] * scale : 0.0f;
    }

    // ---- softmax over the 9 taps (registers only) ----
    float m = s[0];
    #pragma unroll
    for (int j = 1; j < KW; ++j) m = fmaxf(m, s[j]);
    float p[KW];
    float sum = 0.0f;
    #pragma unroll
    for (int j = 0; j < KW; ++j) { p[j] = __expf(s[j] - m); sum += p[j]; }
    const float inv = 1.0f / sum;
    #pragma unroll
    for (int j = 0; j < KW; ++j) {
        bool ok = (unsigned)(gb + j * DIL) < (unsigned)N_TOK;
        p[j] = ok ? p[j] * inv : 0.0f;   // OOB v contributes zero
    }

    // ---- out[dd] = sum_j p[j] * v[dd][tid + 3j]; stage in LDS ----
    #pragma unroll 8
    for (int dd = 0; dd < HD; ++dd) {
        const float* vrow = &vT[dd * KPITCH + tid];
        float acc = 0.0f;
        #pragma unroll
        for (int j = 0; j < KW; ++j)
            acc = fmaf(p[j], vrow[j * DIL], acc);
        oT[tid * OPITCH + dd] = acc;
    }
    __syncthreads();

    // ---- async LDS -> global store: token rows are 32 contiguous floats ----
    // 128 tokens x 8 float4-chunks = 1024 chunk-stores, 8 per thread.
    #pragma unroll
    for (int it = 0; it < 8; ++it) {
        int f     = tid + it * THREADS;     // 0..1023
        int token = f >> 3;
        int ch    = f & 7;
        unsigned ldo  = ldsO + (unsigned)((token * OPITCH + ch * 4) * 4);
        unsigned voff = (unsigned)(((n0 + token) * NCH + ch * 4) * 4);
        asm volatile("global_store_async_from_lds_b128 %0, %1, %2"
                     :: "v"(voff), "v"(ldo), "s"(obase) : "memory");
    }
    asm volatile("s_wait_asynccnt 0x0" ::: "memory");
}

extern "C" void kernel_launch(void* const* d_in, const int* in_sizes, int n_in,
                              void* d_out, int out_size, void* d_ws, size_t ws_size,
                              hipStream_t stream) {
    const float* q = (const float*)d_in[0];
    const float* k = (const float*)d_in[1];
    const float* v = (const float*)d_in[2];
    float* out = (float*)d_out;
    (void)in_sizes; (void)n_in; (void)out_size; (void)d_ws; (void)ws_size;

    dim3 grid(4 * NHEAD * (N_TOK / TILE));   // 2048 blocks
    dim3 block(THREADS);
    hipLaunchKernelGGL(dilated_attn_kernel, grid, block, 0, stream, q, k, v, out);
}